// RecursiveNonlinearEquationSolver_56332791054351
// MI455X (gfx1250) — compile-verified
//
#include <hip/hip_runtime.h>
#include <math.h>

// ---------------------------------------------------------------------------
// RecursiveNonlinearEquationSolver for MI455X (gfx1250, wave32).
//
//   y0:     (1024, 32)  f32
//   A:      (32, 32)    f32
//   forces: (257,1024,32) f32
//   out:    (257,1024,32) f32
//
// Strategy: J1 = I + dt*A*diag(sech^2(y)) with ||dt*A*diag|| ~ 0.02, so
// J1^{-1} u is computed by a 4-term Neumann series (error O(3e-9)), turning
// every dense 32x32 solve into matvecs with A^T.  All matvecs are
// (16 batch x 32) @ (32 x 32) tiles done with V_WMMA_F32_16X16X4_F32
// (full fp32, matching the reference precision): 2 N-tiles x 8 K-steps.
// One wave per workgroup owns 16 batch rows and walks the whole recursion
// (16 chunks x 6 Newton x 16 Thomas steps) with state held in LDS (~46 KB).
// ---------------------------------------------------------------------------

typedef float v2f __attribute__((ext_vector_type(2)));
typedef float v8f __attribute__((ext_vector_type(8)));

#define DT_STEP   0.01f
#define NS        32      // nstate
#define BLK       16      // block size (steps per chunk)
#define NEWTON_IT 6
#define NEUMANN   4       // J^{-1} Neumann terms: err ~ (0.02)^5

#if defined(__AMDGCN__) && __has_builtin(__builtin_amdgcn_wmma_f32_16x16x4_f32)
#define USE_WMMA 1
#else
#define USE_WMMA 0
#endif

// D(16x32) += Amat(16x32) @ A^T, Amat rows sourced from LDS.
//   MODE 0: Amat = T                 (residual GEMM:  A @ tanh(y))
//   MODE 1: Amat = (1 - T^2) * W     (Jacobian off-diag application M*w)
// d0 = N-tile 0 (states 0..15), d1 = N-tile 1 (states 16..31).
template <int MODE>
__device__ __forceinline__ void gemm_AT(const float (&T)[16][NS],
                                        const float (&W)[16][NS],
                                        const v2f (&bf)[2][8],
                                        const float* sA,
                                        int lid, int half,
                                        v8f& d0, v8f& d1)
{
#if USE_WMMA
  #pragma unroll
  for (int kk = 0; kk < 8; ++kk) {
    const int c = kk * 4 + 2 * half;      // 32-bit A-frag: v0=K0/K2, v1=K1/K3
    v2f a;
    if (MODE == 0) {
      a.x = T[lid][c];
      a.y = T[lid][c + 1];
    } else {
      const float t0 = T[lid][c], t1 = T[lid][c + 1];
      a.x = (1.0f - t0 * t0) * W[lid][c];
      a.y = (1.0f - t1 * t1) * W[lid][c + 1];
    }
    d0 = __builtin_amdgcn_wmma_f32_16x16x4_f32(false, a, false, bf[0][kk],
                                               (short)0, d0, false, false);
    d1 = __builtin_amdgcn_wmma_f32_16x16x4_f32(false, a, false, bf[1][kk],
                                               (short)0, d1, false, false);
  }
#else
  // Scalar fallback (kept compilable for safety; same math, same layout).
  #pragma unroll
  for (int r = 0; r < 8; ++r) {
    const int m = r + 8 * half;
    float acc0 = d0[r], acc1 = d1[r];
    #pragma unroll
    for (int c = 0; c < NS; ++c) {
      const float tv = T[m][c];
      const float av = (MODE == 0) ? tv : (1.0f - tv * tv) * W[m][c];
      acc0 += av * sA[(lid     ) * NS + c];   // A^T[c][n] = A[n][c]
      acc1 += av * sA[(lid + 16) * NS + c];
    }
    d0[r] = acc0;
    d1[r] = acc1;
  }
#endif
}

__global__ __launch_bounds__(32)
void rnes_solver(const float* __restrict__ y0,
                 const float* __restrict__ A,
                 const float* __restrict__ forces,
                 float* __restrict__ out,
                 int nbatch, int nchunks)
{
  __shared__ float sY[BLK][16][NS];   // 32 KB: chunk iterate y
  __shared__ float sT[16][NS];        // tanh(y_k) for current step
  __shared__ float sU[16][NS];        // u = R_k + x_{k-1}
  __shared__ float sW[16][NS];        // Neumann iterate -> x_k
  __shared__ float sYold[16][NS];     // pre-update y_{k} (ylag for k+1)
  __shared__ float sYprev[16][NS];    // final state of previous chunk
  __shared__ float sA[NS * NS];       // A row-major

  const int lane = threadIdx.x;       // blockDim.x == 32 (one wave)
  const int lid  = lane & 15;
  const int half = lane >> 4;
  const int b0   = blockIdx.x * 16;   // batch tile base

  for (int i = lane; i < NS * NS; i += 32) sA[i] = A[i];

  // Hoist B fragments of A^T (loop invariant): bf[nt][kk] covers
  // K rows kk*4..kk*4+3, N cols nt*16 + lid.  Layout mirrors the 32-bit
  // fragment pattern: v0 = K0 (lanes 0-15) / K2 (lanes 16-31), v1 = K1/K3.
  v2f bf[2][8];
  #pragma unroll
  for (int nt = 0; nt < 2; ++nt) {
    const int n = nt * 16 + lid;
    #pragma unroll
    for (int kk = 0; kk < 8; ++kk) {
      const int c = kk * 4 + 2 * half;
      v2f b;
      b.x = sA[n * NS + c];       // A^T[c][n]   = A[n][c]
      b.y = sA[n * NS + c + 1];   // A^T[c+1][n] = A[n][c+1]
      bf[nt][kk] = b;
    }
  }

  // yprev <- y0 slice; out row 0 <- y0.
  for (int i = lane; i < 16 * NS; i += 32) {
    const int m = i >> 5, n = i & 31;
    const float v = y0[(size_t)(b0 + m) * NS + n];
    sYprev[m][n] = v;
    out[(size_t)(b0 + m) * NS + n] = v;
  }

  #pragma unroll 1
  for (int c = 0; c < nchunks; ++c) {
    // ZeroPredictor: initial guess 0 for the whole chunk.
    for (int i = lane; i < BLK * 16 * NS; i += 32) ((float*)sY)[i] = 0.0f;

    #pragma unroll 1
    for (int it = 0; it < NEWTON_IT; ++it) {
      // x_{-1} = 0 for the Thomas forward substitution.
      for (int i = lane; i < 16 * NS; i += 32) ((float*)sW)[i] = 0.0f;

      #pragma unroll 1
      for (int k = 0; k < BLK; ++k) {
        const float* fk =
            forces + ((size_t)(c * BLK + k + 1) * nbatch + b0) * NS;
        if (k + 1 < BLK)  // pull next step's forces toward L2 (global_prefetch)
          __builtin_prefetch(fk + (size_t)nbatch * NS, 0, 1);

        // t = tanh(y_k)
        #pragma unroll
        for (int i = lane; i < 16 * NS; i += 32) {
          const int m = i >> 5, n = i & 31;
          sT[m][n] = tanhf(sY[k][m][n]);
        }

        // G = T @ A^T  (fp32 WMMA, C starts at 0)
        v8f g0 = {}, g1 = {};
        gemm_AT<0>(sT, sW, bf, sA, lid, half, g0, g1);

        // u = R_k + x_{k-1};  R = y - ylag + dt*(G - f)
        #pragma unroll
        for (int nt = 0; nt < 2; ++nt) {
          #pragma unroll
          for (int r = 0; r < 8; ++r) {
            const int m = r + 8 * half, n = lid + 16 * nt;
            const float ylag = (k == 0) ? sYprev[m][n] : sYold[m][n];
            const float gv = nt ? g1[r] : g0[r];
            const float R =
                sY[k][m][n] - ylag + DT_STEP * (gv - fk[m * NS + n]);
            sU[m][n] = R + sW[m][n];
          }
        }
        // w = u (Neumann term 0)
        #pragma unroll
        for (int i = lane; i < 16 * NS; i += 32)
          ((float*)sW)[i] = ((const float*)sU)[i];

        // x_k = (I - M + M^2 - M^3 + M^4) u  via  w <- u - M w,
        // M v = dt * (sech^2(y_k) .* v) @ A^T
        #pragma unroll 1
        for (int j = 0; j < NEUMANN; ++j) {
          v8f p0 = {}, p1 = {};
          gemm_AT<1>(sT, sW, bf, sA, lid, half, p0, p1);
          #pragma unroll
          for (int nt = 0; nt < 2; ++nt) {
            #pragma unroll
            for (int r = 0; r < 8; ++r) {
              const int m = r + 8 * half, n = lid + 16 * nt;
              sW[m][n] = sU[m][n] - DT_STEP * (nt ? p1[r] : p0[r]);
            }
          }
        }

        // Save pre-update y_k as next step's ylag, then y_k -= x_k.
        #pragma unroll
        for (int nt = 0; nt < 2; ++nt) {
          #pragma unroll
          for (int r = 0; r < 8; ++r) {
            const int m = r + 8 * half, n = lid + 16 * nt;
            const float yo = sY[k][m][n];
            sYold[m][n] = yo;
            sY[k][m][n] = yo - sW[m][n];
          }
        }
      } // k
    } // newton

    // Carry: yprev = y[last]; emit chunk results.
    for (int i = lane; i < 16 * NS; i += 32) {
      const int m = i >> 5, n = i & 31;
      sYprev[m][n] = sY[BLK - 1][m][n];
    }
    #pragma unroll 1
    for (int k = 0; k < BLK; ++k) {
      for (int i = lane; i < 16 * NS; i += 32) {
        const int m = i >> 5, n = i & 31;
        out[((size_t)(1 + c * BLK + k) * nbatch + (b0 + m)) * NS + n] =
            sY[k][m][n];
      }
    }
  } // chunk
}

extern "C" void kernel_launch(void* const* d_in, const int* in_sizes, int n_in,
                              void* d_out, int out_size, void* d_ws,
                              size_t ws_size, hipStream_t stream) {
  (void)n_in; (void)out_size; (void)d_ws; (void)ws_size;
  const float* y0     = (const float*)d_in[0];
  const float* A      = (const float*)d_in[1];
  const float* forces = (const float*)d_in[2];
  float* out          = (float*)d_out;

  const int nbatch  = in_sizes[0] / NS;            // 1024
  const int n       = in_sizes[2] / (nbatch * NS); // 257
  const int nchunks = (n - 1) / BLK;               // 16

  dim3 grid(nbatch / 16);
  dim3 block(32);
  hipLaunchKernelGGL(rnes_solver, grid, block, 0, stream,
                     y0, A, forces, out, nbatch, nchunks);
}